// NeighborhoodAttention2D_16398185136397
// MI455X (gfx1250) — compile-verified
//
#include <hip/hip_runtime.h>
#include <hip/hip_bf16.h>

// ---------------------------------------------------------------------------
// NeighborhoodAttention2D for MI455X (gfx1250, wave32)
//   B=2 H=64 W=64 C=128 NH=4 HD=32 K=7
//   Stage 1: qkv = x @ w_qkv + b_qkv          (TDM->LDS + fp32 WMMA 16x16x4)
//   Stage 2: per-pixel 7x7 neighborhood attn  (VALU, online softmax)
//   Stage 3: out = attn @ w_proj + b_proj     (TDM->LDS + fp32 WMMA 16x16x4)
// Memory-bound (~35MB vs 23.3 TB/s => ~1.5us floor); fp32 WMMA keeps full
// accuracy at zero cost since we are nowhere near the matrix-op ceiling.
// ---------------------------------------------------------------------------

#define BB 2
#define HH 64
#define WW 64
#define CC 128
#define NHEAD 4
#define HD 32
#define KWIN 7
#define NPIX (BB * HH * WW)          // 8192
#define QKV_N (3 * CC)               // 384
#define SCALE 0.17677669529663687f   // 1/sqrt(32)

typedef __attribute__((ext_vector_type(2))) float        v2f;
typedef __attribute__((ext_vector_type(8))) float        v8f;
typedef __attribute__((ext_vector_type(4))) unsigned int v4u;
typedef __attribute__((ext_vector_type(8))) int          v8i;
typedef __attribute__((ext_vector_type(4))) int          v4i;

// LDS row stride for the staged A tile: 129 dwords (TDM pads 1 dword per
// 128-dword row) -> bank = (row + k) % 64, no conflicts on column reads.
#define LDSA_STRIDE 129

// ---------------------------------------------------------------------------
// Block = 8 waves; all waves share one 16-row M tile (tilesN % 8 == 0 for both
// GEMMs), each wave owns one 16-col N tile.  The 16x128 fp32 A tile is staged
// to LDS by the Tensor Data Mover (async, TENSORcnt), then each wave runs 32
// V_WMMA_F32_16X16X4_F32 accumulations reading A fragments from LDS
// (ds_load_b64) and B fragments from L2-resident weights.
// Fragment layouts (ISA 7.12.2):
//   A 16x4 : VGPR0={lanes0-15:K=0, lanes16-31:K=2}, VGPR1={K=1,K=3}, M=lane%16
//   B 4x16 : VGPR0={row K=0 lanes0-15, row K=2 lanes16-31}, VGPR1={K=1,K=3}
//   C 16x16: VGPR g = row (g + 8*(lane>=16)), col lane%16
// ---------------------------------------------------------------------------
__global__ void __launch_bounds__(256)
wmma_gemm_bias_f32(const float* __restrict__ A,
                   const float* __restrict__ Bm,
                   const float* __restrict__ bias,
                   float* __restrict__ C,
                   int M, int N)
{
    __shared__ float ldsA[16 * LDSA_STRIDE];   // 8256 B of the 320KB WGP LDS

    const int wib    = threadIdx.x >> 5;       // wave in block, 0..7
    const int lane   = threadIdx.x & 31;
    const int tilesN = N >> 4;
    const int tile   = blockIdx.x * 8 + wib;
    const int tm     = (tile / tilesN) << 4;   // shared by all waves in block
    const int tn     = (tile % tilesN) << 4;

    // ---- TDM: stage A[tm:tm+16, 0:128] into LDS (wave 0 issues) -----------
    if (threadIdx.x < 32) {
        const int tm0 = ((blockIdx.x * 8) / tilesN) << 4;     // == tm, uniform
        const unsigned long long gaddr =
            (unsigned long long)(size_t)(A + (size_t)tm0 * CC);
        const unsigned ldsOff = (unsigned)(size_t)(&ldsA[0]); // addr[31:0] = LDS offset

        v4u g0;
        g0[0] = 1u;                                        // count=1, user desc
        g0[1] = ldsOff;                                    // lds_addr
        g0[2] = (unsigned)(gaddr & 0xFFFFFFFFu);           // global_addr[31:0]
        g0[3] = (unsigned)((gaddr >> 32) & 0x01FFFFFFu)    // global_addr[56:32]
              | (2u << 30);                                // type=2 (image)

        v8i g1;
        g1[0] = (2 << 16)        // data_size = 4 bytes
              | (1 << 20)        // pad_enable
              | (6 << 22);       // pad_interval = 128 dwords; pad_amount=0 (1 dw)
        g1[1] = (CC & 0xFFFF) << 16;         // tensor_dim0 = 128 (K extent)
        g1[2] = (M  & 0xFFFF) << 16;         // tensor_dim1 = M rows
        g1[3] = (CC & 0xFFFF) << 16;         // tile_dim0 = 128
        g1[4] = 16;                          // tile_dim1 = 16 rows, tile_dim2=0
        g1[5] = CC;                          // tensor_dim0_stride = 128 elems
        g1[6] = 0;
        g1[7] = 0;

        const v4i gz  = {0, 0, 0, 0};        // 2D tensor: groups 2/3 unused
        const v8i gz8 = {0, 0, 0, 0, 0, 0, 0, 0};
        __builtin_amdgcn_tensor_load_to_lds(g0, g1, gz, gz, gz8, 0);
        __builtin_amdgcn_s_wait_tensorcnt((short)0);
    }
    __syncthreads();

    // ---- WMMA accumulation -------------------------------------------------
    const int half = lane >> 4;              // 0: K pair {0,1}, 1: {2,3}
    const int l16  = lane & 15;

    v8f acc = {};
    const float* arow = &ldsA[l16 * LDSA_STRIDE];
    const float* bcol = Bm + tn + l16;

    #pragma unroll 8
    for (int k = 0; k < CC; k += 4) {
        const int ka = k + half * 2;
        v2f af, bf;
        af.x = arow[ka];                     // ds_load_b64 (consecutive pair)
        af.y = arow[ka + 1];
        bf.x = bcol[(size_t)ka * N];
        bf.y = bcol[(size_t)(ka + 1) * N];
        acc = __builtin_amdgcn_wmma_f32_16x16x4_f32(
            /*neg_a=*/false, af, /*neg_b=*/false, bf,
            /*c_mod=*/(short)0, acc, /*reuse_a=*/false, /*reuse_b=*/false);
    }

    const float bv = bias[tn + l16];
    #pragma unroll
    for (int g = 0; g < 8; ++g) {
        const int row = tm + g + half * 8;
        C[(size_t)row * N + tn + l16] = acc[g] + bv;
    }
}

// ---------------------------------------------------------------------------
// Neighborhood attention: one thread per (pixel, head).
// qkv layout per pixel: [q(0..127) | k(128..255) | v(256..383)], head slice n*32.
// Flash-style online softmax over the 49 in-window neighbors; qkv rows are
// L2-resident (12.6MB << 192MB) so the 49x re-reads never touch HBM.
// ---------------------------------------------------------------------------
__global__ void __launch_bounds__(256)
natten2d_kernel(const float* __restrict__ qkv,
                const float* __restrict__ rpb,
                float* __restrict__ out)
{
    const int idx = blockIdx.x * blockDim.x + threadIdx.x;
    if (idx >= NPIX * NHEAD) return;

    const int n    = idx & (NHEAD - 1);
    const int pix  = idx >> 2;                 // b*H*W + h*W + w
    const int w    = pix % WW;
    const int h    = (pix / WW) % HH;
    const int bimg = pix / (HH * WW);

    // clipped window starts (matches jnp.clip(i - K//2, 0, L - K))
    int sh = h - (KWIN / 2); sh = sh < 0 ? 0 : sh; sh = sh > HH - KWIN ? HH - KWIN : sh;
    int sw = w - (KWIN / 2); sw = sw < 0 ? 0 : sw; sw = sw > WW - KWIN ? WW - KWIN : sw;

    // load q (pre-scaled)
    const float4* qp = reinterpret_cast<const float4*>(qkv + (size_t)pix * QKV_N + n * HD);
    float q[HD];
    #pragma unroll
    for (int d4 = 0; d4 < HD / 4; ++d4) {
        float4 t = qp[d4];
        q[4 * d4 + 0] = t.x * SCALE;
        q[4 * d4 + 1] = t.y * SCALE;
        q[4 * d4 + 2] = t.z * SCALE;
        q[4 * d4 + 3] = t.w * SCALE;
    }

    const float* rpb_n = rpb + n * (2 * KWIN - 1) * (2 * KWIN - 1);

    float m = -3.0e38f;
    float l = 0.0f;
    float acc[HD];
    #pragma unroll
    for (int d = 0; d < HD; ++d) acc[d] = 0.0f;

    for (int p = 0; p < KWIN; ++p) {
        const int hh  = sh + p;
        const int bih = hh - h + (KWIN - 1);   // 0..12
        for (int qq = 0; qq < KWIN; ++qq) {
            const int ww2 = sw + qq;
            const int biw = ww2 - w + (KWIN - 1);
            const size_t nbr = (size_t)((bimg * HH + hh) * WW + ww2) * QKV_N;

            const float4* kp = reinterpret_cast<const float4*>(qkv + nbr + CC + n * HD);
            const float4* vp = reinterpret_cast<const float4*>(qkv + nbr + 2 * CC + n * HD);

            float s = rpb_n[bih * (2 * KWIN - 1) + biw];
            #pragma unroll
            for (int d4 = 0; d4 < HD / 4; ++d4) {
                float4 kv = kp[d4];
                s += q[4 * d4 + 0] * kv.x + q[4 * d4 + 1] * kv.y
                   + q[4 * d4 + 2] * kv.z + q[4 * d4 + 3] * kv.w;
            }

            const float mn   = fmaxf(m, s);
            const float cold = __expf(m - mn);   // 0 on first iter
            const float pnew = __expf(s - mn);
            l = l * cold + pnew;
            #pragma unroll
            for (int d4 = 0; d4 < HD / 4; ++d4) {
                float4 vv = vp[d4];
                acc[4 * d4 + 0] = acc[4 * d4 + 0] * cold + pnew * vv.x;
                acc[4 * d4 + 1] = acc[4 * d4 + 1] * cold + pnew * vv.y;
                acc[4 * d4 + 2] = acc[4 * d4 + 2] * cold + pnew * vv.z;
                acc[4 * d4 + 3] = acc[4 * d4 + 3] * cold + pnew * vv.w;
            }
            m = mn;
        }
    }

    const float inv_l = 1.0f / l;
    float* op = out + (size_t)pix * CC + n * HD;
    #pragma unroll
    for (int d = 0; d < HD; ++d) op[d] = acc[d] * inv_l;
}

// ---------------------------------------------------------------------------
extern "C" void kernel_launch(void* const* d_in, const int* in_sizes, int n_in,
                              void* d_out, int out_size, void* d_ws, size_t ws_size,
                              hipStream_t stream)
{
    (void)in_sizes; (void)n_in; (void)out_size; (void)ws_size;

    const float* x      = (const float*)d_in[0];
    const float* w_qkv  = (const float*)d_in[1];
    const float* b_qkv  = (const float*)d_in[2];
    const float* rpb    = (const float*)d_in[3];
    const float* w_proj = (const float*)d_in[4];
    const float* b_proj = (const float*)d_in[5];
    float* out = (float*)d_out;

    float* qkv  = (float*)d_ws;                       // NPIX * 384 floats (12.6 MB)
    float* attn = qkv + (size_t)NPIX * QKV_N;         // NPIX * 128 floats (4.2 MB)

    // Stage 1: qkv = x @ w_qkv + b_qkv   (M=8192, N=384, K=128)
    {
        const int tiles  = (NPIX / 16) * (QKV_N / 16);    // 12288 waves
        const int blocks = tiles / 8;                     // 8 waves/block
        wmma_gemm_bias_f32<<<blocks, 256, 0, stream>>>(
            x, w_qkv, b_qkv, qkv, NPIX, QKV_N);
    }

    // Stage 2: neighborhood attention   (one thread per pixel*head)
    {
        const int total = NPIX * NHEAD;                   // 32768
        natten2d_kernel<<<total / 256, 256, 0, stream>>>(qkv, rpb, attn);
    }

    // Stage 3: out = attn @ w_proj + b_proj  (M=8192, N=128, K=128)
    {
        const int tiles  = (NPIX / 16) * (CC / 16);       // 4096 waves
        const int blocks = tiles / 8;                     // 512
        wmma_gemm_bias_f32<<<blocks, 256, 0, stream>>>(
            attn, w_proj, b_proj, out, NPIX, CC);
    }
}